// GraphScoringLayer_31310311588447
// MI455X (gfx1250) — compile-verified
//
#include <hip/hip_runtime.h>

// ---------------------------------------------------------------------------
// RGCN scoring: scores = relu(x@rootW + sum_r mean_r@W_r + bias) @ linW + linB
// B=8, N=4096, C=128, R=16, E=65536
// fp32 throughout, matrix math via V_WMMA_F32_16X16X4_F32 (CDNA5 wave32).
// ---------------------------------------------------------------------------

typedef __attribute__((ext_vector_type(2))) float v2f;
typedef __attribute__((ext_vector_type(8))) float v8f;

#define Bq   8
#define Nq   4096
#define Cq   128
#define Rq   16
#define Eq   65536
#define BNq  (Bq * Nq)   // 32768 nodes
#define BEq  (Bq * Eq)   // 524288 edges

// Packed-W LDS pitch: one row per k-pair, element (k,n) at
//   (k>>1)*PITCH + 2n + (k&1)
// => a B fragment {W[kk][n], W[kk+1][n]} is one aligned b64 load.
// PITCH = 2*128 + 32 makes lane-halves (k-pair e vs e+1) bank-disjoint:
//   lanes 0-15 -> banks {2n % 64}      = 0..31
//   lanes16-31 -> banks {(32+2n) % 64} = 32..63
#define PITCHq 288

// ---------------------------------------------------------------------------
// Zero the workspace (agg + cnt), float4 grid-stride stores.
// ---------------------------------------------------------------------------
__global__ void zero_ws_kernel(float4* __restrict__ w, int n4) {
  int i = blockIdx.x * blockDim.x + threadIdx.x;
  int stride = gridDim.x * blockDim.x;
  float4 z = {0.0f, 0.0f, 0.0f, 0.0f};
  for (; i < n4; i += stride) w[i] = z;
}

// ---------------------------------------------------------------------------
// Per-(node, relation) in-degree counts. Float atomics: counts <= 65536 are
// exactly representable in fp32.
// ---------------------------------------------------------------------------
__global__ void count_kernel(const int* __restrict__ ei,
                             const int* __restrict__ et,
                             float* __restrict__ cnt) {
  int e = blockIdx.x * blockDim.x + threadIdx.x;
  if (e >= BEq) return;
  int b  = e >> 16;          // E == 65536
  int el = e & (Eq - 1);
  int dst = ei[(b * 2 + 1) * Eq + el];
  int r   = et[e];
  int gdst = b * Nq + dst;
  atomicAdd(&cnt[gdst * Rq + r], 1.0f);
}

// ---------------------------------------------------------------------------
// One wave32 per edge: agg[dst][r][:] += x[src][:] / cnt[dst][r]
// (division folded in, so agg holds the per-relation means directly).
// Each lane handles channels lane, lane+32, lane+64, lane+96.
// ---------------------------------------------------------------------------
__global__ void scatter_kernel(const float* __restrict__ x,
                               const int*   __restrict__ ei,
                               const int*   __restrict__ et,
                               const float* __restrict__ cnt,
                               float*       __restrict__ agg) {
  int wave = threadIdx.x >> 5;
  int lane = threadIdx.x & 31;
  int e = blockIdx.x * 8 + wave;          // grid sized exactly: BEq/8 blocks
  int b  = e >> 16;
  int el = e & (Eq - 1);
  int src = ei[(b * 2 + 0) * Eq + el];
  int dst = ei[(b * 2 + 1) * Eq + el];
  int r   = et[e];
  int gsrc = b * Nq + src;
  int gdst = b * Nq + dst;
  float inv = 1.0f / cnt[gdst * Rq + r];  // >= 1 since this edge is in the bin
  const float* xb = x   + (size_t)gsrc * Cq;
  float*       ab = agg + ((size_t)gdst * Rq + r) * Cq;
#pragma unroll
  for (int cc = 0; cc < Cq; cc += 32) {
    atomicAdd(&ab[lane + cc], xb[lane + cc] * inv);
  }
}

// ---------------------------------------------------------------------------
// Fused GEMM + ReLU + linear scoring.
//   block = 256 threads = 8 waves; block tile = 128 rows x 128 cols.
//   wave tile = 16 rows x 128 cols = 8 WMMA C-tiles (v8f each).
//   K-superloop: rblk 0 = x @ rootW, rblk 1..16 = mean_{r} @ W_{r}.
//   Each 128x128 fp32 weight block staged in LDS (fragment-packed, 72 KB).
// A 16x4 fragment (ISA layout): lanes 0-15 -> K, K+1 ; lanes 16-31 -> K+2, K+3
// B 4x16 fragment: v0 = row K (lanes 0-15) / row K+2 (16-31), v1 = K+1 / K+3
// C 16x16: VGPR j = row j (lanes 0-15) / row j+8 (lanes 16-31), N = lane&15
// ---------------------------------------------------------------------------
__global__ void __launch_bounds__(256)
rgcn_gemm_score_kernel(const float* __restrict__ x,
                       const float* __restrict__ agg,
                       const float* __restrict__ rootW,
                       const float* __restrict__ relW,
                       const float* __restrict__ bias,
                       const float* __restrict__ linW,
                       const float* __restrict__ linB,
                       float* __restrict__ scores) {
  __shared__ float lds_w[(Cq / 2) * PITCHq];   // 64 k-pairs * 288 = 72 KB

  const int tid   = threadIdx.x;
  const int wave  = tid >> 5;
  const int lane  = tid & 31;
  const int lhalf = lane >> 4;       // 0: lanes 0-15, 1: lanes 16-31
  const int l15   = lane & 15;
  const int m0    = blockIdx.x * 128 + wave * 16;

  // Accumulators init with rgcn_bias (per output column).
  v8f acc[8];
#pragma unroll
  for (int t = 0; t < 8; ++t) {
    float bv = bias[t * 16 + l15];
#pragma unroll
    for (int j = 0; j < 8; ++j) acc[t][j] = bv;
  }

  for (int rblk = 0; rblk < 17; ++rblk) {
    const float* Wsrc = (rblk == 0) ? rootW
                                    : (relW + (size_t)(rblk - 1) * Cq * Cq);
    __syncthreads();
    // Stage W into fragment-packed LDS layout: (k,n) -> (k>>1)*P + 2n + (k&1)
    for (int i = tid * 4; i < Cq * Cq; i += 256 * 4) {
      const int k = i >> 7;
      const int n = i & 127;
      const float4 w = *(const float4*)&Wsrc[i];
      float* p = &lds_w[(k >> 1) * PITCHq + (k & 1) + 2 * n];
      p[0] = w.x; p[2] = w.y; p[4] = w.z; p[6] = w.w;
    }
    __syncthreads();

    const int arow = m0 + l15;
    const float* Abase = (rblk == 0)
        ? (x   + (size_t)arow * Cq)
        : (agg + ((size_t)arow * Rq + (rblk - 1)) * Cq);

#pragma unroll 2
    for (int k = 0; k < Cq; k += 4) {
      const int kk = k + 2 * lhalf;
      const float2 av = *(const float2*)(Abase + kk);  // 8B aligned (kk even)
      v2f a; a.x = av.x; a.y = av.y;
      const int kp = (k >> 1) + lhalf;                 // k-pair row in LDS
      const float* brow = &lds_w[kp * PITCHq];
#pragma unroll
      for (int t = 0; t < 8; ++t) {
        const int n = t * 16 + l15;
        const v2f bf = *(const v2f*)&brow[2 * n];      // one b64, no repack
        acc[t] = __builtin_amdgcn_wmma_f32_16x16x4_f32(
            false, a, false, bf, (short)0, acc[t], false, false);
      }
    }
  }

  // Fused: relu(out) @ linW + linB, reduced across the 16-lane halves
  // matching the WMMA C layout. Writes 16 rows per wave.
  float p[8];
#pragma unroll
  for (int j = 0; j < 8; ++j) p[j] = 0.0f;
#pragma unroll
  for (int t = 0; t < 8; ++t) {
    float lw = linW[t * 16 + l15];
#pragma unroll
    for (int j = 0; j < 8; ++j) {
      float v = acc[t][j];
      v = v > 0.0f ? v : 0.0f;
      p[j] += v * lw;
    }
  }
  float lb = linB[0];
#pragma unroll
  for (int j = 0; j < 8; ++j) {
#pragma unroll
    for (int off = 8; off >= 1; off >>= 1)
      p[j] += __shfl_xor(p[j], off, 16);
    if (l15 == 0) scores[m0 + 8 * lhalf + j] = p[j] + lb;
  }
}

// ---------------------------------------------------------------------------
// Host launcher. Workspace layout:
//   agg : BN * R * C floats (268.4 MB)  -- per-(node, relation) mean features
//   cnt : BN * R     floats (  2.1 MB)  -- per-(node, relation) in-degree
// ---------------------------------------------------------------------------
extern "C" void kernel_launch(void* const* d_in, const int* in_sizes, int n_in,
                              void* d_out, int out_size, void* d_ws, size_t ws_size,
                              hipStream_t stream) {
  (void)in_sizes; (void)n_in; (void)out_size; (void)ws_size;

  const float* x     = (const float*)d_in[0];  // node_features (B,N,C)
  const int*   ei    = (const int*)  d_in[1];  // edge_index    (B,2,E)
  const int*   et    = (const int*)  d_in[2];  // edge_type     (B,E)
  const float* relW  = (const float*)d_in[3];  // rel_weight    (R,C,C)
  const float* rootW = (const float*)d_in[4];  // root_weight   (C,C)
  const float* bias  = (const float*)d_in[5];  // rgcn_bias     (C,)
  const float* linW  = (const float*)d_in[6];  // lin_weight    (C,1)
  const float* linB  = (const float*)d_in[7];  // lin_bias      (1,)
  float* scores = (float*)d_out;               // (B,N)

  float* agg = (float*)d_ws;
  float* cnt = agg + (size_t)BNq * Rq * Cq;

  const int totalF = BNq * Rq * Cq + BNq * Rq;   // 67,633,152 (divisible by 4)
  zero_ws_kernel<<<2048, 256, 0, stream>>>((float4*)d_ws, totalF / 4);
  count_kernel<<<BEq / 256, 256, 0, stream>>>(ei, et, cnt);
  scatter_kernel<<<BEq / 8, 256, 0, stream>>>(x, ei, et, cnt, agg);
  rgcn_gemm_score_kernel<<<BNq / 128, 256, 0, stream>>>(
      x, agg, rootW, relW, bias, linW, linB, scores);
}